// ExpandMask_10685878632847
// MI455X (gfx1250) — compile-verified
//
#include <hip/hip_runtime.h>
#include <cstdint>

// ExpandMask: y[2i] = x[i-1]+x[i]+x[i+1] ; y[2i+1] = x[i]+x[i+1] ; out = (y > 0.5) as float
// Bandwidth-bound (~198 MB @ 23.3 TB/s => ~8.5us floor). CDNA5 path:
// async global->LDS staging (ASYNCcnt) + conflict-free ds_load_b128 + NT b128 streaming stores.

typedef __attribute__((ext_vector_type(4))) float v4f;

#define L_ROW   262144   // per-row length of the input mask (TILE divides L_ROW)
#define TILE    2048     // inputs per block
#define HALF    1024     // two 4-element groups per thread: li and li+HALF
#define THREADS 256      // 8 wave32 waves; 8 inputs -> 16 outputs per thread

// Stencil for 4 consecutive inputs x0..x3 with neighbors xm1 (left) / x4 (right):
//   o_k = x_k + x_{k+1}            (odd outputs, j = 2i+1)
//   e_k = x_{k-1} + o_k            (even outputs, j = 2i)
__device__ __forceinline__ void stencil4(float xm1, v4f x, float x4,
                                         v4f& r0, v4f& r1)
{
    const float o0 = x.x + x.y;
    const float o1 = x.y + x.z;
    const float o2 = x.z + x.w;
    const float o3 = x.w + x4;
    const float e0 = xm1 + o0;
    const float e1 = x.x + o1;
    const float e2 = x.y + o2;
    const float e3 = x.z + o3;
    r0 = (v4f){ e0 > 0.5f ? 1.0f : 0.0f,
                o0 > 0.5f ? 1.0f : 0.0f,
                e1 > 0.5f ? 1.0f : 0.0f,
                o1 > 0.5f ? 1.0f : 0.0f };
    r1 = (v4f){ e2 > 0.5f ? 1.0f : 0.0f,
                o2 > 0.5f ? 1.0f : 0.0f,
                e3 > 0.5f ? 1.0f : 0.0f,
                o3 > 0.5f ? 1.0f : 0.0f };
}

__global__ __launch_bounds__(THREADS) void expand_mask_kernel(
    const float* __restrict__ xin, float* __restrict__ out)
{
    __shared__ float s[TILE];

    const int tid = threadIdx.x;
    const int li  = tid * 4;                                  // group0 local index
    const long long tile0 = (long long)blockIdx.x * TILE;
    const long long gi0   = tile0 + li;                       // group0 global index
    const long long gi1   = gi0 + HALF;                       // group1 global index

    // ---- Async copies: 2 x 16 B per lane, global -> LDS (ASYNCcnt) ----
    // VDST vgpr = LDS byte address (low 32 bits of generic shared ptr),
    // VADDR vgpr pair = 64-bit global address, GV mode.
    unsigned     lds0 = (unsigned)(uintptr_t)(&s[li]);
    unsigned     lds1 = (unsigned)(uintptr_t)(&s[li + HALF]);
    const float* g0   = xin + gi0;
    const float* g1   = xin + gi1;
    asm volatile("global_load_async_to_lds_b128 %0, %1, off"
                 :: "v"(lds0), "v"(g0) : "memory");
    asm volatile("global_load_async_to_lds_b128 %0, %1, off"
                 :: "v"(lds1), "v"(g1) : "memory");

    // ---- Halo taps: only the tile-edge lanes touch global again (row-masked) ----
    float haloL = 0.0f, haloR = 0.0f;
    if (li == 0) {                                            // tid == 0
        if ((tile0 & (L_ROW - 1)) != 0)
            haloL = xin[gi0 - 1];
    }
    if (li + 4 == HALF) {                                     // tid == 255
        if (((tile0 + TILE) & (L_ROW - 1)) != 0)
            haloR = xin[gi1 + 4];
    }

    // Drain this wave's async copies, then block-barrier before any LDS read.
    asm volatile("s_wait_asynccnt 0" ::: "memory");
    __syncthreads();

    // ---- Conflict-free LDS reads (tid*16 B stride -> each b128 spans all banks once) ----
    const v4f xa = *(const v4f*)&s[li];
    const v4f xb = *(const v4f*)&s[li + HALF];
    const float xm1a = (li == 0) ? haloL : s[li - 1];
    const float x4a  = s[li + 4];                 // always in LDS: s[HALF] = group1's first elem
    const float xm1b = s[li + HALF - 1];          // always in LDS: s[HALF-1] = group0's last elem
    const float x4b  = (li + 4 == HALF) ? haloR : s[li + HALF + 4];

    v4f a0, a1, b0, b1;
    stencil4(xm1a, xa, x4a, a0, a1);
    stencil4(xm1b, xb, x4b, b0, b1);

    // ---- Streaming NT b128 stores: out flat index = 2*gi .. 2*gi+7 (32 B aligned) ----
    v4f* oa = (v4f*)(out + 2 * gi0);
    __builtin_nontemporal_store(a0, oa);
    __builtin_nontemporal_store(a1, oa + 1);
    v4f* ob = (v4f*)(out + 2 * gi1);
    __builtin_nontemporal_store(b0, ob);
    __builtin_nontemporal_store(b1, ob + 1);
}

extern "C" void kernel_launch(void* const* d_in, const int* in_sizes, int n_in,
                              void* d_out, int out_size, void* d_ws, size_t ws_size,
                              hipStream_t stream) {
    (void)n_in; (void)d_ws; (void)ws_size; (void)out_size;
    const float* x   = (const float*)d_in[0];
    float*       out = (float*)d_out;
    const int total_in = in_sizes[0];        // 64 * 262144 = 16,777,216
    const int blocks   = total_in / TILE;    // 8,192 blocks
    expand_mask_kernel<<<blocks, THREADS, 0, stream>>>(x, out);
}